// LSTM_Regression_83846351552960
// MI455X (gfx1250) — compile-verified
//
#include <hip/hip_runtime.h>
#include <math.h>

// LSTM regression, MI455X (gfx1250, wave32).
// One lane = one batch element; one wave = 32 elements = two 16-row M-tiles.
// Recurrent gate GEMMs run on the matrix pipe via V_WMMA_F32_16X16X4_F32.
// All nonlinearities use the CDNA5 hardware V_TANH_F32 (TRANS pipe, which
// co-executes with XDL WMMA): sigmoid(x) = 0.5*tanh(x/2)+0.5. Branch-free,
// so EXEC stays all-ones for WMMA and no OCML range-reduction bloat.
//
// Roofline: 176 MB of x reads -> 7.6 us HBM floor @ 23.3 TB/s; ~4.6 GFLOP
// fp32 (85% in the recurrent GEMMs -> matrix pipe) + ~260M tanh (TRANS pipe).

#define BLOCK 256
#define TSEQ  14
#define FIN   12
#define HID   7
#define NG    28           // 4*H gates, order i,f,g,o
#define UPITCH 16          // u-buffer row pitch (floats)
#define ZPITCH 36          // z-buffer row pitch: 144B -> conflict-free b128 reads

typedef __attribute__((ext_vector_type(2))) float v2f;
typedef __attribute__((ext_vector_type(4))) float v4f;
typedef __attribute__((ext_vector_type(8))) float v8f;

__device__ __forceinline__ void lds_fence() {
  // Wave-local ordering: DS ops from one wave execute in order (ISA 7.3);
  // we only need to stop the compiler from reordering LDS accesses.
  __builtin_amdgcn_wave_barrier();
}

// Hardware tanh (TRANS pipe). ISA 7.4: one independent op / V_NOP required
// after a TRANS op before its output is consumed — the builtin models this;
// the asm fallback carries its own v_nop.
__device__ __forceinline__ float tanh_hw(float x) {
#if __has_builtin(__builtin_amdgcn_tanhf)
  return __builtin_amdgcn_tanhf(x);
#else
  float r;
  asm volatile("v_tanh_f32 %0, %1\n\tv_nop" : "=v"(r) : "v"(x));
  return r;
#endif
}

__device__ __forceinline__ float sigmoid_hw(float x) {
  return fmaf(0.5f, tanh_hw(0.5f * x), 0.5f);
}

// One layer's gate GEMM for the whole wave:
//   z[32 x 28] = u[32 x 4*KCH] * Wt[4*KCH x 28] + bias
// A layout (32-bit 16x4, ISA 7.12.2): lane l holds row M=l%16, K = 2*(l/16)+{0,1}
// C layout: VGPR r, lanes 0-15 -> (M=r, N=l), lanes 16-31 -> (M=r+8, N=l-16)
// B layout symmetric to C/D row-split: VGPR0 = K{0,2}, VGPR1 = K{1,3}.
template <int KCH>
__device__ __forceinline__ void gemm_tile(const float* uw, float* zw,
                                          const v2f (&Bw)[KCH][2],
                                          const v8f& Clo, const v8f& Chi,
                                          int nl, int half) {
#pragma unroll
  for (int m = 0; m < 2; ++m) {              // two 16-row M tiles per wave
    v8f C0 = Clo, C1 = Chi;                  // bias pre-loaded in C octets
#pragma unroll
    for (int c = 0; c < KCH; ++c) {          // K chunks of 4
      v2f A = *(const v2f*)(uw + (16 * m + nl) * UPITCH + 4 * c + 2 * half);
      C0 = __builtin_amdgcn_wmma_f32_16x16x4_f32(false, A, false, Bw[c][0],
                                                 (short)0, C0, false, false);
      C1 = __builtin_amdgcn_wmma_f32_16x16x4_f32(false, A, false, Bw[c][1],
                                                 (short)0, C1, false, false);
    }
#pragma unroll
    for (int r = 0; r < 8; ++r) {            // scatter C tile -> z rows
      const int row = 16 * m + r + 8 * half;
      zw[row * ZPITCH + nl]      = C0[r];
      zw[row * ZPITCH + 16 + nl] = C1[r];
    }
  }
}

// Per-lane LSTM cell update from this lane's 28 gate pre-activations.
// Branch-free: 5 v_tanh + ~9 VALU per hidden unit.
__device__ __forceinline__ void cell_update(const float* zr, float* h, float* c) {
  float g[NG];
#pragma unroll
  for (int q = 0; q < 7; ++q) {              // 7 x b128 conflict-free reads
    v4f v = *(const v4f*)(zr + 4 * q);
    g[4 * q + 0] = v.x; g[4 * q + 1] = v.y;
    g[4 * q + 2] = v.z; g[4 * q + 3] = v.w;
  }
#pragma unroll
  for (int j = 0; j < HID; ++j) {
    const float ig = sigmoid_hw(g[j]);
    const float fg = sigmoid_hw(g[HID + j]);
    const float gg = tanh_hw(g[2 * HID + j]);
    const float og = sigmoid_hw(g[3 * HID + j]);
    const float cc = fg * c[j] + ig * gg;
    c[j] = cc;
    h[j] = og * tanh_hw(cc);
  }
}

__global__ __launch_bounds__(BLOCK) void lstm_reg_kernel(
    const float* __restrict__ x,
    const float* __restrict__ w1, const float* __restrict__ b1,
    const float* __restrict__ w_ih0, const float* __restrict__ w_hh0,
    const float* __restrict__ b_ih0, const float* __restrict__ b_hh0,
    const float* __restrict__ w_ih1, const float* __restrict__ w_hh1,
    const float* __restrict__ b_ih1, const float* __restrict__ b_hh1,
    const float* __restrict__ w2, const float* __restrict__ b2,
    float* __restrict__ out) {
  // Packed transposed weights, shared by all waves (built once per block):
  // layer0: Wt0[k][n], k=0 -> w_ih0[n], k=1..7 -> w_hh0[n][k-1]
  // layer1: Wt1[k][n], k=0..6 -> w_ih1[n][k], k=7..13 -> w_hh1[n][k-7]
  __shared__ float wt0[8][32];
  __shared__ float wt1[16][32];
  __shared__ float bia0[32];
  __shared__ float bia1[32];
  // Per-wave transpose scratch (no cross-wave sharing, no block barriers in loop)
  __shared__ float u_lds[BLOCK / 32][32][UPITCH];
  __shared__ float z_lds[BLOCK / 32][32][ZPITCH];

  const int tid  = threadIdx.x;
  const int lane = tid & 31;
  const int wv   = tid >> 5;
  const int nl   = lane & 15;   // N / M-row index within tile
  const int half = lane >> 4;   // K-pair select for A/B operands
  const long b   = (long)blockIdx.x * BLOCK + tid;   // batch element (B % 256 == 0)

  // ---- stage packed weights into LDS ----
  {
    int i = tid;                               // 8*32 = 256 entries
    const int k = i >> 5, n = i & 31;
    float v = 0.f;
    if (n < NG) {
      if (k == 0) v = w_ih0[n];
      else if (k <= HID) v = w_hh0[n * HID + (k - 1)];
    }
    wt0[k][n] = v;
  }
  for (int i = tid; i < 16 * 32; i += BLOCK) { // 512 entries
    const int k = i >> 5, n = i & 31;
    float v = 0.f;
    if (n < NG) {
      if (k < HID) v = w_ih1[n * HID + k];
      else if (k < 2 * HID) v = w_hh1[n * HID + (k - HID)];
    }
    wt1[k][n] = v;
  }
  if (tid < 32) {
    bia0[tid] = (tid < NG) ? (b_ih0[tid] + b_hh0[tid]) : 0.f;
    bia1[tid] = (tid < NG) ? (b_ih1[tid] + b_hh1[tid]) : 0.f;
  }
  __syncthreads();

  // ---- phase 1: s[t] = tanh(x[b,t,:] . w1 + b1) ----
  float w1r[FIN];
#pragma unroll
  for (int f = 0; f < FIN; ++f) w1r[f] = w1[f];   // uniform -> scalar loads
  const float b1r = b1[0];
  float s[TSEQ];
  const float* xb = x + b * (TSEQ * FIN);         // 672B per lane, 16B aligned
#pragma unroll
  for (int t = 0; t < TSEQ; ++t) {
    const v4f a0 = *(const v4f*)(xb + t * FIN + 0);
    const v4f a1 = *(const v4f*)(xb + t * FIN + 4);
    const v4f a2 = *(const v4f*)(xb + t * FIN + 8);
    float acc = b1r;
    acc += a0.x * w1r[0]; acc += a0.y * w1r[1];
    acc += a0.z * w1r[2]; acc += a0.w * w1r[3];
    acc += a1.x * w1r[4]; acc += a1.y * w1r[5];
    acc += a1.z * w1r[6]; acc += a1.w * w1r[7];
    acc += a2.x * w1r[8]; acc += a2.y * w1r[9];
    acc += a2.z * w1r[10]; acc += a2.w * w1r[11];
    s[t] = tanh_hw(acc);
  }

  // ---- build per-wave B operands once (weights live in 24 VGPRs) ----
  v2f B0[2][2];  // [k-chunk][n-tile], layer 0 (K=8)
  v2f B1[4][2];  // layer 1 (K=16, top 2 rows zero)
#pragma unroll
  for (int c = 0; c < 2; ++c)
#pragma unroll
    for (int v = 0; v < 2; ++v) {
      const int k0 = 4 * c + 2 * half, n = 16 * v + nl;
      v2f t; t.x = wt0[k0][n]; t.y = wt0[k0 + 1][n];
      B0[c][v] = t;
    }
#pragma unroll
  for (int c = 0; c < 4; ++c)
#pragma unroll
    for (int v = 0; v < 2; ++v) {
      const int k0 = 4 * c + 2 * half, n = 16 * v + nl;
      v2f t; t.x = wt1[k0][n]; t.y = wt1[k0 + 1][n];
      B1[c][v] = t;
    }
  // bias octets kept live in registers as the WMMA C seed
  v8f Cb00, Cb01, Cb10, Cb11;
  {
    const float b00 = bia0[nl], b01 = bia0[16 + nl];
    const float b10 = bia1[nl], b11 = bia1[16 + nl];
#pragma unroll
    for (int r = 0; r < 8; ++r) {
      Cb00[r] = b00; Cb01[r] = b01; Cb10[r] = b10; Cb11[r] = b11;
    }
  }

  // ---- recurrent state in registers ----
  float h0[HID] = {}, c0[HID] = {}, h1[HID] = {}, c1[HID] = {};
  float* uw = &u_lds[wv][0][0];
  float* zw = &z_lds[wv][0][0];
  v4f* up = (v4f*)(uw + lane * UPITCH);

  // Serial dependence through (h,c): unrolling buys no ILP, only I$ misses.
#pragma unroll 1
  for (int t = 0; t < TSEQ; ++t) {
    // ---- layer 0: u = [s_t, h0(7)] (K=8) ----
    {
      v4f u0; u0.x = s[t];  u0.y = h0[0]; u0.z = h0[1]; u0.w = h0[2];
      v4f u1; u1.x = h0[3]; u1.y = h0[4]; u1.z = h0[5]; u1.w = h0[6];
      up[0] = u0; up[1] = u1;
    }
    lds_fence();
    gemm_tile<2>(uw, zw, B0, Cb00, Cb01, nl, half);
    lds_fence();
    cell_update(zw + lane * ZPITCH, h0, c0);
    lds_fence();
    // ---- layer 1: u = [h0_new(7), h1_prev(7), 0, 0] (K=16) ----
    {
      v4f u0; u0.x = h0[0]; u0.y = h0[1]; u0.z = h0[2]; u0.w = h0[3];
      v4f u1; u1.x = h0[4]; u1.y = h0[5]; u1.z = h0[6]; u1.w = h1[0];
      v4f u2; u2.x = h1[1]; u2.y = h1[2]; u2.z = h1[3]; u2.w = h1[4];
      v4f u3; u3.x = h1[5]; u3.y = h1[6]; u3.z = 0.f;   u3.w = 0.f;
      up[0] = u0; up[1] = u1; up[2] = u2; up[3] = u3;
    }
    lds_fence();
    gemm_tile<4>(uw, zw, B1, Cb10, Cb11, nl, half);
    lds_fence();
    cell_update(zw + lane * ZPITCH, h1, c1);
    lds_fence();
  }

  // ---- final linear: out = [h0, h1] . w2 + b2 ----
  float w2r[2 * HID];
#pragma unroll
  for (int j = 0; j < 2 * HID; ++j) w2r[j] = w2[j];
  float acc = b2[0];
#pragma unroll
  for (int j = 0; j < HID; ++j) acc += w2r[j] * h0[j];
#pragma unroll
  for (int j = 0; j < HID; ++j) acc += w2r[HID + j] * h1[j];
  out[b] = acc;
}

extern "C" void kernel_launch(void* const* d_in, const int* in_sizes, int n_in,
                              void* d_out, int out_size, void* d_ws, size_t ws_size,
                              hipStream_t stream) {
  (void)n_in; (void)d_ws; (void)ws_size; (void)out_size;
  const float* x     = (const float*)d_in[0];
  const float* w1    = (const float*)d_in[1];
  const float* b1    = (const float*)d_in[2];
  const float* w_ih0 = (const float*)d_in[3];
  const float* w_hh0 = (const float*)d_in[4];
  const float* b_ih0 = (const float*)d_in[5];
  const float* b_hh0 = (const float*)d_in[6];
  const float* w_ih1 = (const float*)d_in[7];
  const float* w_hh1 = (const float*)d_in[8];
  const float* b_ih1 = (const float*)d_in[9];
  const float* b_hh1 = (const float*)d_in[10];
  const float* w2    = (const float*)d_in[11];
  const float* b2    = (const float*)d_in[12];
  float* out = (float*)d_out;

  const int B = in_sizes[0] / (TSEQ * FIN);   // 262144, divisible by BLOCK
  const int grid = B / BLOCK;
  lstm_reg_kernel<<<grid, BLOCK, 0, stream>>>(x, w1, b1, w_ih0, w_hh0, b_ih0,
                                              b_hh0, w_ih1, w_hh1, b_ih1, b_hh1,
                                              w2, b2, out);
}